// GriddingSample_37873021616739
// MI455X (gfx1250) — compile-verified
//
#include <hip/hip_runtime.h>
#include <stdint.h>

// GriddingSample: 8-corner trilinear gather from a [B, 128^3] grid.
// B=16, N=32768, G=128 (G^3 = 2^21). Memory-latency bound; grid fits in L2.

__global__ __launch_bounds__(256) void gridding_sample_kernel(
    const float* __restrict__ grid,   // [B * 2^21]
    const float* __restrict__ pt,     // [B * N * 3]
    float* __restrict__ out,          // [B * N]
    int total)                        // B * N
{
  // 256 points/block -> 3072 B of coordinates, staged via CDNA5 async
  // global->LDS b128 copies. Each wave stages its own 384 B slice
  // (24 lanes x 16 B), so only a per-wave s_wait_asynccnt is needed.
  __shared__ __align__(16) float scoord[256 * 3];

  const int tid  = threadIdx.x;
  const int lane = tid & 31;
  const int w    = tid >> 5;
  const int pid  = blockIdx.x * 256 + tid;

  {
    unsigned ldsoff = (unsigned)(uintptr_t)(
        (__attribute__((address_space(3))) char*)scoord + (w * 384 + lane * 16));
    uint64_t gaddr = (uint64_t)(uintptr_t)(
        (const char*)pt + (size_t)blockIdx.x * 3072 + (size_t)w * 384 + (size_t)lane * 16);
    if (lane < 24) {
      // ASYNCcnt-tracked DMA of 16 B/lane into LDS (coalesced, bypasses VGPRs)
      asm volatile("global_load_async_to_lds_b128 %0, %1, off"
                   :: "v"(ldsoff), "v"(gaddr)
                   : "memory");
    }
    asm volatile("s_wait_asynccnt 0" ::: "memory");
  }

  // Read this thread's (x,y,z) from LDS: stride-3 words across lanes,
  // gcd(3,64)=1 -> bank-conflict free.
  const float* c = &scoord[tid * 3];
  float px = c[0] + 64.0f;
  float py = c[1] + 64.0f;
  float pz = c[2] + 64.0f;

  float lxf = floorf(px), lyf = floorf(py), lzf = floorf(pz);
  float fx = px - lxf, fy = py - lyf, fz = pz - lzf;
  int ix = (int)lxf, iy = (int)lyf, iz = (int)lzf;

  // reference: clip(lo+d, 0, G-1) for each corner
  int ix0 = min(max(ix,     0), 127), ix1 = min(max(ix + 1, 0), 127);
  int iy0 = min(max(iy,     0), 127), iy1 = min(max(iy + 1, 0), 127);
  int iz0 = min(max(iz,     0), 127), iz1 = min(max(iz + 1, 0), 127);

  // batch = pid / 32768 ; per-batch grid = 2^21 floats
  const float* gb = grid + ((size_t)(pid >> 15) << 21);

  int r00 = (ix0 << 14) + (iy0 << 7);
  int r01 = (ix0 << 14) + (iy1 << 7);
  int r10 = (ix1 << 14) + (iy0 << 7);
  int r11 = (ix1 << 14) + (iy1 << 7);

  // Issue all 8 independent gathers up front; wave32 keeps them in flight.
  float v000 = gb[r00 + iz0], v001 = gb[r00 + iz1];
  float v010 = gb[r01 + iz0], v011 = gb[r01 + iz1];
  float v100 = gb[r10 + iz0], v101 = gb[r10 + iz1];
  float v110 = gb[r11 + iz0], v111 = gb[r11 + iz1];

  float wx1 = fx, wx0 = 1.0f - fx;
  float wy1 = fy, wy0 = 1.0f - fy;
  float wz1 = fz, wz0 = 1.0f - fz;

  float c00 = fmaf(wz1, v001, wz0 * v000);
  float c01 = fmaf(wz1, v011, wz0 * v010);
  float c10 = fmaf(wz1, v101, wz0 * v100);
  float c11 = fmaf(wz1, v111, wz0 * v110);
  float cy0 = fmaf(wy1, c01, wy0 * c00);
  float cy1 = fmaf(wy1, c11, wy0 * c10);
  float res = fmaf(wx1, cy1, wx0 * cy0);

  if (pid < total) out[pid] = res;
}

extern "C" void kernel_launch(void* const* d_in, const int* in_sizes, int n_in,
                              void* d_out, int out_size, void* d_ws, size_t ws_size,
                              hipStream_t stream) {
  const float* grid = (const float*)d_in[0];   // [16, 128^3] f32
  const float* pt   = (const float*)d_in[1];   // [16, 32768, 3] f32
  float* out = (float*)d_out;                  // [16, 32768] f32
  (void)in_sizes; (void)n_in; (void)d_ws; (void)ws_size;

  int total  = out_size;                 // 524288
  int blocks = (total + 255) / 256;      // 2048 blocks of 8 wave32s
  gridding_sample_kernel<<<blocks, 256, 0, stream>>>(grid, pt, out, total);
}